// AliBiAttention_20547123544618
// MI455X (gfx1250) — compile-verified
//
#include <hip/hip_runtime.h>
#include <hip/hip_bf16.h>

#define HID   1024
#define NHEAD 16
#define DH    64
#define BATCH 2
#define SEQ   2048
#define MROWS (BATCH*SEQ)

typedef __attribute__((ext_vector_type(2))) float    v2f;
typedef __attribute__((ext_vector_type(8))) float    v8f;
typedef __attribute__((ext_vector_type(4))) unsigned v4u;
typedef __attribute__((ext_vector_type(8))) int      v8i;
typedef __attribute__((ext_vector_type(4))) int      v4i;

#if __has_builtin(__builtin_amdgcn_tensor_load_to_lds) && __has_builtin(__builtin_amdgcn_s_wait_tensorcnt)
#define USE_TDM 1
#else
#define USE_TDM 0
#endif

__device__ __forceinline__ v8f wmma4(v2f a, v2f b, v8f c) {
    // D = A(16x4,f32) * B(4x16,f32) + C(16x16,f32)
    return __builtin_amdgcn_wmma_f32_16x16x4_f32(
        false, a, false, b, (short)0, c, false, false);
}

#if USE_TDM
// ---------------------------------------------------------------------------
// Tensor Data Mover: DMA a 2D fp32 tile (tile_d0 contiguous elems x tile_d1
// rows, row stride stride_elts) from global memory into LDS, inserting
// pad_amt(+1) DWORDs of padding after every 2^(pad_int+1) DWORDs so the LDS
// rows land on a bank-conflict-free stride. D# packed per CDNA5 ISA §8.
// 6-arg builtin form (clang-23 / therock-10.0 headers).
// ---------------------------------------------------------------------------
__device__ __forceinline__ void tdm_load_2d(
    const void* gsrc, unsigned lds_off,
    unsigned tile_d0, unsigned tile_d1, unsigned long long stride_elts,
    unsigned pad_int, unsigned pad_amt)
{
    const unsigned long long ga = (unsigned long long)(size_t)gsrc;
    v4u g0;
    g0.x = 1u;                                            // count=1, user descriptor
    g0.y = lds_off;                                       // lds_addr (bytes)
    g0.z = (unsigned)(ga & 0xFFFFFFFFu);                  // global_addr[31:0]
    g0.w = (unsigned)((ga >> 32) & 0x01FFFFFFu) | (2u << 30); // addr[56:32] | type=2
    v8i g1;
    g1[0] = (int)((2u << 16)                              // data_size = 4B
                | (1u << 20)                              // pad_enable
                | (pad_int << 22) | (pad_amt << 25));
    g1[1] = (int)((tile_d0 & 0xFFFFu) << 16);             // tensor_dim0[15:0]
    g1[2] = (int)(((tile_d0 >> 16) & 0xFFFFu) | ((tile_d1 & 0xFFFFu) << 16));
    g1[3] = (int)(((tile_d1 >> 16) & 0xFFFFu) | ((tile_d0 & 0xFFFFu) << 16)); // tile_dim0
    g1[4] = (int)(tile_d1 & 0xFFFFu);                     // tile_dim1 | tile_dim2=0
    g1[5] = (int)(unsigned)(stride_elts & 0xFFFFFFFFull); // tensor_dim0_stride lo32
    g1[6] = (int)(unsigned)((stride_elts >> 32) & 0xFFFFull);
    g1[7] = 0;
    const v4i z4 = {0, 0, 0, 0};
    const v8i z8 = {0, 0, 0, 0, 0, 0, 0, 0};
    __builtin_amdgcn_tensor_load_to_lds(g0, g1, z4, z4, z8, 0);
}
#endif

__device__ __forceinline__ unsigned lds_off_of(const void* p) {
    // Generic LDS address: offset lives in the low 32 bits (flat-LDS mapping).
    return (unsigned)(size_t)p;
}

// ---------------------------------------------------------------------------
// Tiled GEMM: C[M,N] = X[M,HID] @ W[HID,N], f32 WMMA, TDM-staged tiles,
// double-buffered LDS. Block tile 64x64, BK=16, 8 waves, wave tile 16x32.
// Epilogue splits columns: [0,HID)->dq, [HID,2HID)->dk, [2HID,3HID)->dv.
// ---------------------------------------------------------------------------
__global__ __launch_bounds__(256) void gemm_f32_kernel(
    const float* __restrict__ X, const float* __restrict__ W, int N,
    float* __restrict__ dq, float* __restrict__ dk, float* __restrict__ dv)
{
    __shared__ float As[2][64][20];   // 16-float rows @ stride 20 (TDM pad 16+4)
    __shared__ float Bs[2][16][68];   // 64-float rows @ stride 68 (TDM pad 64+4)

    const int t    = threadIdx.x;
    const int wave = t >> 5;
    const int lane = t & 31;
    const int half = lane >> 4;
    const int l16  = lane & 15;
    const int wm   = wave >> 1;
    const int wn   = wave & 1;
    const int gm   = blockIdx.y * 64;
    const int gn   = blockIdx.x * 64;
    const int NK   = HID / 16;

    v8f acc0 = {}; v8f acc1 = {};

#if USE_TDM
    if (wave == 0) {
        tdm_load_2d(X + (size_t)gm * HID,        lds_off_of(&As[0][0][0]), 16, 64, HID, 3, 3);
        tdm_load_2d(W + (size_t)gn,              lds_off_of(&Bs[0][0][0]), 64, 16, (unsigned long long)N, 5, 3);
    }
#else
    const int arow = t >> 2,  acg = (t & 3)  * 4;
    const int brow = t >> 4,  bcg = (t & 15) * 4;
    {
        const float4 av = *(const float4*)(X + (size_t)(gm + arow) * HID + acg);
        As[0][arow][acg+0]=av.x; As[0][arow][acg+1]=av.y; As[0][arow][acg+2]=av.z; As[0][arow][acg+3]=av.w;
        const float4 bv = *(const float4*)(W + (size_t)brow * N + gn + bcg);
        Bs[0][brow][bcg+0]=bv.x; Bs[0][brow][bcg+1]=bv.y; Bs[0][brow][bcg+2]=bv.z; Bs[0][brow][bcg+3]=bv.w;
    }
#endif

    for (int i = 0; i < NK; ++i) {
        const int cur = i & 1;
#if USE_TDM
        if (wave == 0) {
            if (i + 1 < NK) {
                tdm_load_2d(X + (size_t)gm * HID + (i + 1) * 16,
                            lds_off_of(&As[cur ^ 1][0][0]), 16, 64, HID, 3, 3);
                tdm_load_2d(W + (size_t)((i + 1) * 16) * N + gn,
                            lds_off_of(&Bs[cur ^ 1][0][0]), 64, 16, (unsigned long long)N, 5, 3);
                __builtin_amdgcn_s_wait_tensorcnt(2);   // tile i complete (in-order)
            } else {
                __builtin_amdgcn_s_wait_tensorcnt(0);
            }
        }
#else
        if (i + 1 < NK) {
            const int k0 = (i + 1) * 16;
            const float4 av = *(const float4*)(X + (size_t)(gm + arow) * HID + k0 + acg);
            As[cur^1][arow][acg+0]=av.x; As[cur^1][arow][acg+1]=av.y; As[cur^1][arow][acg+2]=av.z; As[cur^1][arow][acg+3]=av.w;
            const float4 bv = *(const float4*)(W + (size_t)(k0 + brow) * N + gn + bcg);
            Bs[cur^1][brow][bcg+0]=bv.x; Bs[cur^1][brow][bcg+1]=bv.y; Bs[cur^1][brow][bcg+2]=bv.z; Bs[cur^1][brow][bcg+3]=bv.w;
        }
#endif
        __syncthreads();

        #pragma unroll
        for (int kc = 0; kc < 4; ++kc) {
            const int kk = kc * 4 + half * 2;
            const float2 a2 = *(const float2*)&As[cur][wm*16 + l16][kk];
            v2f a; a.x = a2.x; a.y = a2.y;
            v2f b0; b0.x = Bs[cur][kk][wn*32 + l16];      b0.y = Bs[cur][kk+1][wn*32 + l16];
            v2f b1; b1.x = Bs[cur][kk][wn*32 + 16 + l16]; b1.y = Bs[cur][kk+1][wn*32 + 16 + l16];
            acc0 = wmma4(a, b0, acc0);
            acc1 = wmma4(a, b1, acc1);
        }
        __syncthreads();
    }

    #pragma unroll
    for (int ns = 0; ns < 2; ++ns) {
        v8f acc = ns ? acc1 : acc0;
        #pragma unroll
        for (int r = 0; r < 8; ++r) {
            const int row = gm + wm*16 + r + half*8;
            const int col = gn + wn*32 + ns*16 + l16;
            const float v = acc[r];
            if (col < HID)          dq[(size_t)row*HID + col]         = v;
            else if (col < 2*HID)   dk[(size_t)row*HID + col - HID]   = v;
            else                    dv[(size_t)row*HID + col - 2*HID] = v;
        }
    }
}

// ---------------------------------------------------------------------------
// Flash-style ALiBi attention. Grid: (SEQ/128, BATCH*NHEAD), 256 threads.
// Each wave owns one 16-query tile; K/V 16x64 tiles TDM-staged, double
// buffered. NOTE: reference does NOT scale QK^T by 1/sqrt(dk).
// ---------------------------------------------------------------------------
__global__ __launch_bounds__(256) void attn_kernel(
    const float* __restrict__ Q,  const float* __restrict__ Kc,
    const float* __restrict__ Vc, const int* __restrict__ mask,
    float* __restrict__ attn)
{
    __shared__ float Ks[2][16][68];
    __shared__ float Vs[2][16][68];
    __shared__ float Ps[8][16][18];

    const int t    = threadIdx.x;
    const int wave = t >> 5;
    const int lane = t & 31;
    const int half = lane >> 4;
    const int l16  = lane & 15;
    const int b    = (int)blockIdx.y >> 4;
    const int h    = (int)blockIdx.y & 15;
    const int qbase = (int)blockIdx.x * 128 + wave * 16;
    const float slope = exp2f(-0.5f * (float)(h + 1));   // H=16 => 2^(-8/16*(h+1))
    const int NT = SEQ / 16;

    // Preload Q A-fragments: row m = l16, dims split by lane half per A layout
    v2f aq[16];
    {
        const float* qrow = Q + ((size_t)(b*SEQ + qbase + l16)) * HID + h*DH;
        #pragma unroll
        for (int kc = 0; kc < 16; ++kc) {
            const int kk = kc * 4 + half * 2;
            aq[kc].x = qrow[kk]; aq[kc].y = qrow[kk + 1];
        }
    }

    float mrow[8], lrow[8];
    #pragma unroll
    for (int r = 0; r < 8; ++r) { mrow[r] = -1e30f; lrow[r] = 0.f; }
    v8f o0 = {}, o1 = {}, o2 = {}, o3 = {};

#if USE_TDM
    if (wave == 0) {
        tdm_load_2d(Kc + ((size_t)(b*SEQ))*HID + h*DH, lds_off_of(&Ks[0][0][0]), 64, 16, HID, 5, 3);
        tdm_load_2d(Vc + ((size_t)(b*SEQ))*HID + h*DH, lds_off_of(&Vs[0][0][0]), 64, 16, HID, 5, 3);
    }
#else
    const int ldrow = t >> 4, ldcg = (t & 15) * 4;
    {
        const float4 kv = *(const float4*)(Kc + ((size_t)(b*SEQ + ldrow))*HID + h*DH + ldcg);
        Ks[0][ldrow][ldcg+0]=kv.x; Ks[0][ldrow][ldcg+1]=kv.y; Ks[0][ldrow][ldcg+2]=kv.z; Ks[0][ldrow][ldcg+3]=kv.w;
        const float4 vv = *(const float4*)(Vc + ((size_t)(b*SEQ + ldrow))*HID + h*DH + ldcg);
        Vs[0][ldrow][ldcg+0]=vv.x; Vs[0][ldrow][ldcg+1]=vv.y; Vs[0][ldrow][ldcg+2]=vv.z; Vs[0][ldrow][ldcg+3]=vv.w;
    }
#endif

    for (int kt = 0; kt < NT; ++kt) {
        const int cur = kt & 1;
#if USE_TDM
        if (wave == 0) {
            if (kt + 1 < NT) {
                const size_t nrow = (size_t)(b*SEQ + (kt + 1)*16);
                tdm_load_2d(Kc + nrow*HID + h*DH, lds_off_of(&Ks[cur^1][0][0]), 64, 16, HID, 5, 3);
                tdm_load_2d(Vc + nrow*HID + h*DH, lds_off_of(&Vs[cur^1][0][0]), 64, 16, HID, 5, 3);
                __builtin_amdgcn_s_wait_tensorcnt(2);   // tile kt complete
            } else {
                __builtin_amdgcn_s_wait_tensorcnt(0);
            }
        }
#else
        if (kt + 1 < NT) {
            const int skey = (kt + 1)*16 + ldrow;
            const float4 kv = *(const float4*)(Kc + ((size_t)(b*SEQ + skey))*HID + h*DH + ldcg);
            Ks[cur^1][ldrow][ldcg+0]=kv.x; Ks[cur^1][ldrow][ldcg+1]=kv.y; Ks[cur^1][ldrow][ldcg+2]=kv.z; Ks[cur^1][ldrow][ldcg+3]=kv.w;
            const float4 vv = *(const float4*)(Vc + ((size_t)(b*SEQ + skey))*HID + h*DH + ldcg);
            Vs[cur^1][ldrow][ldcg+0]=vv.x; Vs[cur^1][ldrow][ldcg+1]=vv.y; Vs[cur^1][ldrow][ldcg+2]=vv.z; Vs[cur^1][ldrow][ldcg+3]=vv.w;
        }
#endif
        __syncthreads();

        // S = Q @ K^T  (B[k][n] = K[n][k]; lane's column n = key index l16)
        v8f s = {};
        #pragma unroll
        for (int kc = 0; kc < 16; ++kc) {
            const int kk = kc * 4 + half * 2;
            const float2 k2 = *(const float2*)&Ks[cur][l16][kk];
            v2f bf; bf.x = k2.x; bf.y = k2.y;
            s = wmma4(aq[kc], bf, s);
        }

        const int  j      = kt*16 + l16;                 // key column owned by lane
        const bool masked = (mask[b*SEQ + j] == 0);

        #pragma unroll
        for (int r = 0; r < 8; ++r) {
            const int   i    = qbase + r + half*8;       // query row for this element
            const float bias = slope * fminf((float)(j - i), 0.f);
            const float sc   = masked ? -1e9f : (s[r] + bias);

            float rm = sc;                               // row max across 16-lane half
            rm = fmaxf(rm, __shfl_xor(rm, 1));
            rm = fmaxf(rm, __shfl_xor(rm, 2));
            rm = fmaxf(rm, __shfl_xor(rm, 4));
            rm = fmaxf(rm, __shfl_xor(rm, 8));
            const float mnew  = fmaxf(mrow[r], rm);
            const float alpha = __expf(mrow[r] - mnew);
            const float p     = __expf(sc - mnew);
            float ps = p;
            ps += __shfl_xor(ps, 1); ps += __shfl_xor(ps, 2);
            ps += __shfl_xor(ps, 4); ps += __shfl_xor(ps, 8);
            lrow[r] = lrow[r] * alpha + ps;
            mrow[r] = mnew;
            o0[r] *= alpha; o1[r] *= alpha; o2[r] *= alpha; o3[r] *= alpha;
            Ps[wave][r + half*8][l16] = p;               // C-layout -> LDS
        }
        asm volatile("s_wait_dscnt 0x0" ::: "memory");   // intra-wave cross-lane handoff

        // O += P @ V  (A = P 16x16 from LDS, B = V tile from LDS)
        #pragma unroll
        for (int kc = 0; kc < 4; ++kc) {
            const int kk = kc * 4 + half * 2;
            const float2 p2 = *(const float2*)&Ps[wave][l16][kk];
            v2f a; a.x = p2.x; a.y = p2.y;
            v2f b0; b0.x = Vs[cur][kk][ 0 + l16]; b0.y = Vs[cur][kk+1][ 0 + l16];
            v2f b1; b1.x = Vs[cur][kk][16 + l16]; b1.y = Vs[cur][kk+1][16 + l16];
            v2f b2; b2.x = Vs[cur][kk][32 + l16]; b2.y = Vs[cur][kk+1][32 + l16];
            v2f b3; b3.x = Vs[cur][kk][48 + l16]; b3.y = Vs[cur][kk+1][48 + l16];
            o0 = wmma4(a, b0, o0);
            o1 = wmma4(a, b1, o1);
            o2 = wmma4(a, b2, o2);
            o3 = wmma4(a, b3, o3);
        }
        __syncthreads();
    }

    #pragma unroll
    for (int r = 0; r < 8; ++r) {
        const int   srow = qbase + r + half*8;
        const float inv  = 1.0f / lrow[r];
        float* orow = attn + ((size_t)(b*SEQ + srow)) * HID + h*DH;
        orow[ 0 + l16] = o0[r] * inv;
        orow[16 + l16] = o1[r] * inv;
        orow[32 + l16] = o2[r] * inv;
        orow[48 + l16] = o3[r] * inv;
    }
}

extern "C" void kernel_launch(void* const* d_in, const int* in_sizes, int n_in,
                              void* d_out, int out_size, void* d_ws, size_t ws_size,
                              hipStream_t stream) {
    const float* x    = (const float*)d_in[0];
    const int*   mask = (const int*)  d_in[1];
    const float* Wqkv = (const float*)d_in[2];
    const float* Wout = (const float*)d_in[3];

    float* out    = (float*)d_out;                       // [MROWS,HID]
    float* kcache = out    + (size_t)MROWS * HID;        // [MROWS,HID]
    float* vcache = kcache + (size_t)MROWS * HID;        // [MROWS,HID]
    float* qbuf   = out;                                 // reuse out region as q scratch
    float* attnb  = (float*)d_ws;                        // [MROWS,HID] scratch

    dim3 blk(256);

    // qkv = x @ Wqkv ; q -> qbuf, k -> k_cache, v -> v_cache
    gemm_f32_kernel<<<dim3(3*HID/64, MROWS/64), blk, 0, stream>>>(
        x, Wqkv, 3*HID, qbuf, kcache, vcache);

    // flash attention with ALiBi bias + padding mask
    attn_kernel<<<dim3(SEQ/128, BATCH*NHEAD), blk, 0, stream>>>(
        qbuf, kcache, vcache, mask, attnb);

    // out = attn @ Wout  (overwrites q scratch region with final output)
    gemm_f32_kernel<<<dim3(HID/64, MROWS/64), blk, 0, stream>>>(
        attnb, Wout, HID, out, out, out);
}